// K_means_clustering_45286135169450
// MI455X (gfx1250) — compile-verified
//
#include <hip/hip_runtime.h>

typedef __attribute__((ext_vector_type(2))) float v2f;
typedef __attribute__((ext_vector_type(8))) float v8f;

#define KPROB 13248   // B*S = 64*207
#define NPTS  32
#define DDIM  64
#define CCL   8
#define CPAD  16      // centers padded to 16 rows; rows 8..15 stay zero
#define NITER 15

__global__ __launch_bounds__(32) void kmeans_lloyd_kernel(
    const float* __restrict__ gfeat,
    float* __restrict__ out_id,       // KPROB*NPTS floats
    float* __restrict__ out_centers,  // KPROB*CCL*DDIM floats
    float* __restrict__ ws1,          // per-problem sum of min_d2
    float* __restrict__ ws2)          // per-problem sum of (frac-1/C)^2
{
    __shared__ float sf[NPTS][DDIM];   // features (row-major, point x dim)
    __shared__ float sc[CPAD][DDIM];   // centers (rows 8..15 always zero)
    __shared__ float sfc[NPTS][16];    // f @ centers^T (cols >= 8 unused)
    __shared__ float sf2[NPTS];
    __shared__ float sc2[CCL];
    __shared__ float scnt[CCL];
    __shared__ int   sidx[NPTS];
    __shared__ float sred[NPTS];

    const int lane = threadIdx.x;      // 0..31, one wave32
    const int half = lane >> 4;        // 0 or 1 (lane group)
    const int lm   = lane & 15;
    const int p    = blockIdx.x;       // problem id

    // ---- load features for this problem (coalesced float4) ----
    const float4* g4  = (const float4*)gfeat + (size_t)p * (NPTS * DDIM / 4);
    float4* sf4 = (float4*)&sf[0][0];
    float4* sc4 = (float4*)&sc[0][0];
#pragma unroll
    for (int i = 0; i < 16; ++i) sf4[i * 32 + lane] = g4[i * 32 + lane];
    __syncthreads();

    // ---- init centers = f[:8,:]; zero pad rows 8..15; per-point norms ----
    const float4 z4 = make_float4(0.f, 0.f, 0.f, 0.f);
#pragma unroll
    for (int i = 0; i < 4; ++i) {
        sc4[i * 32 + lane]       = sf4[i * 32 + lane];  // rows 0..7
        sc4[128 + i * 32 + lane] = z4;                  // rows 8..15 = 0
    }
    {
        const float4* r = (const float4*)&sf[lane][0];
        float a = 0.f;
#pragma unroll
        for (int i = 0; i < DDIM / 4; ++i) {
            float4 x = r[i];
            a += x.x * x.x + x.y * x.y + x.z * x.z + x.w * x.w;
        }
        sf2[lane] = a;
    }
    __syncthreads();

    int final_id = 0;

    for (int it = 0; it <= NITER; ++it) {
        // ---- c2[c] = ||center_c||^2 (lanes 0..7) ----
        if (lane < CCL) {
            const float4* r = (const float4*)&sc[lane][0];
            float a = 0.f;
#pragma unroll
            for (int i = 0; i < DDIM / 4; ++i) {
                float4 x = r[i];
                a += x.x * x.x + x.y * x.y + x.z * x.z + x.w * x.w;
            }
            sc2[lane] = a;
        }
        __syncthreads();

        // ---- fc = f(32x64) @ centers^T(64x8) via V_WMMA_F32_16X16X4_F32 ----
        // A 16x4 layout: lane<16 holds M=lane K={4k,4k+1}; lane>=16 K={4k+2,4k+3}
        // B 4x16 mirrored: N=lm, same K pair per lane half; rows 8..15 of sc are 0
#pragma unroll
        for (int mt = 0; mt < 2; ++mt) {
            v8f acc = {0.f, 0.f, 0.f, 0.f, 0.f, 0.f, 0.f, 0.f};
#pragma unroll
            for (int k = 0; k < 16; ++k) {
                const int col = 4 * k + 2 * half;
                v2f a, b;
                a.x = sf[mt * 16 + lm][col];
                a.y = sf[mt * 16 + lm][col + 1];
                b.x = sc[lm][col];          // unconditional: padded rows are 0
                b.y = sc[lm][col + 1];
                acc = __builtin_amdgcn_wmma_f32_16x16x4_f32(
                    false, a, false, b, (short)0, acc, false, false);
            }
            // D layout: lane group half, VGPR v -> M = v + 8*half, N = lm
#pragma unroll
            for (int v = 0; v < 8; ++v)
                sfc[mt * 16 + 8 * half + v][lm] = acc[v];
        }
        __syncthreads();

        // ---- per-point argmin over d2 = f2 - 2 fc + c2 (first-min ties) ----
        const float f2l = sf2[lane];
        float best = 3.4e38f;
        int   bi   = 0;
#pragma unroll
        for (int c = 0; c < CCL; ++c) {
            float d2 = f2l - 2.f * sfc[lane][c] + sc2[c];
            if (d2 < best) { best = d2; bi = c; }
        }
        sidx[lane] = bi;
        if (lane < CCL) scnt[lane] = 0.f;
        __syncthreads();
        atomicAdd(&scnt[bi], 1.0f);   // ds_add_f32
        __syncthreads();

        if (it == NITER) {            // final assignment pass: no update
            final_id = bi;
            sred[lane] = best;
            break;
        }

        // ---- sums = onehot^T(8x32) @ f(32x64) via WMMA; update centers ----
#pragma unroll
        for (int nt = 0; nt < 4; ++nt) {
            v8f acc = {0.f, 0.f, 0.f, 0.f, 0.f, 0.f, 0.f, 0.f};
#pragma unroll
            for (int k = 0; k < 8; ++k) {
                const int pp = 4 * k + 2 * half;
                const int i0 = sidx[pp];
                const int i1 = sidx[pp + 1];
                v2f a, b;
                a.x = (i0 == lm) ? 1.f : 0.f;   // rows m>=8 auto-zero (idx<8)
                a.y = (i1 == lm) ? 1.f : 0.f;
                b.x = sf[pp][nt * 16 + lm];
                b.y = sf[pp + 1][nt * 16 + lm];
                acc = __builtin_amdgcn_wmma_f32_16x16x4_f32(
                    false, a, false, b, (short)0, acc, false, false);
            }
            // valid rows m=0..7 live in lanes 0..15, VGPR v = center c
            if (half == 0) {
#pragma unroll
                for (int v = 0; v < 8; ++v) {
                    float cnt = scnt[v];
                    if (cnt > 0.f) sc[v][nt * 16 + lm] = acc[v] / cnt;
                }
            }
        }
        __syncthreads();
    }

    // ---- outputs ----
    out_id[(size_t)p * NPTS + lane] = (float)final_id;

    float4* oc4 = (float4*)(out_centers + (size_t)p * (CCL * DDIM));
#pragma unroll
    for (int i = 0; i < 4; ++i) oc4[i * 32 + lane] = sc4[i * 32 + lane];

    __syncthreads();
    if (lane == 0) {
        float s = 0.f;
#pragma unroll
        for (int l = 0; l < NPTS; ++l) s += sred[l];
        ws1[p] = s;
        float u = 0.f;
#pragma unroll
        for (int c = 0; c < CCL; ++c) {
            float fr = scnt[c] * (1.0f / (float)NPTS) - (1.0f / (float)CCL);
            u += fr * fr;
        }
        ws2[p] = u;
    }
}

// Deterministic final reduction (fixed order; no float global atomics).
__global__ __launch_bounds__(256) void kmeans_loss_reduce(
    const float* __restrict__ ws1, const float* __restrict__ ws2,
    float* __restrict__ out_loss)
{
    __shared__ float s1[256];
    __shared__ float s2[256];
    const int t = threadIdx.x;
    float a = 0.f, b = 0.f;
    for (int i = t; i < KPROB; i += 256) { a += ws1[i]; b += ws2[i]; }
    s1[t] = a; s2[t] = b;
    __syncthreads();
    for (int off = 128; off > 0; off >>= 1) {
        if (t < off) { s1[t] += s1[t + off]; s2[t] += s2[t + off]; }
        __syncthreads();
    }
    if (t == 0) {
        out_loss[0] = s1[0] / (float)((size_t)KPROB * NPTS);
        out_loss[1] = s2[0] / (float)((size_t)KPROB * CCL);
    }
}

extern "C" void kernel_launch(void* const* d_in, const int* in_sizes, int n_in,
                              void* d_out, int out_size, void* d_ws, size_t ws_size,
                              hipStream_t stream) {
    (void)in_sizes; (void)n_in; (void)out_size; (void)ws_size;
    const float* feat = (const float*)d_in[0];
    float* out         = (float*)d_out;
    float* out_id      = out;                                      // KPROB*NPTS
    float* out_centers = out + (size_t)KPROB * NPTS;               // KPROB*CCL*DDIM
    float* out_loss    = out_centers + (size_t)KPROB * CCL * DDIM; // 2 floats
    float* ws1 = (float*)d_ws;
    float* ws2 = ws1 + KPROB;

    kmeans_lloyd_kernel<<<KPROB, 32, 0, stream>>>(feat, out_id, out_centers, ws1, ws2);
    kmeans_loss_reduce<<<1, 256, 0, stream>>>(ws1, ws2, out_loss);
}